// VNPointNet_31765578121806
// MI455X (gfx1250) — compile-verified
//
#include <hip/hip_runtime.h>
#include <math.h>

// ---------------- problem constants (match reference setup_inputs) ----------
#define BB   4
#define NN   4096
#define KK   20
#define MTOT (BB * 3 * NN)      // 49152, GEMM column dim, m = (b*3+v)*NN + n
#define EPSV  1e-6f
#define BNEPS 1e-5f

typedef __attribute__((ext_vector_type(2))) float v2f;
typedef __attribute__((ext_vector_type(8))) float v8f;

// ============================ utility ======================================
__global__ void zero_f32(float* __restrict__ p, int n) {
    int i = blockIdx.x * 256 + threadIdx.x;
    if (i < n) p[i] = 0.f;
}

// ============================ 1. kNN =======================================
// One wave32 per query point (b,i). Each lane keeps a sorted top-20 list in
// LDS over its strided slice of the 4096 candidates, then a 20-round k-way
// merge across the wave via shfl/ballot. Self (dist 0) is included, matching
// jax.lax.top_k on -||xi-xj||^2.
__global__ void knn_kernel(const float* __restrict__ x, int* __restrict__ idxOut) {
    __shared__ float sd[8 * 32 * KK];
    __shared__ int   si[8 * 32 * KK];
    const int lane = threadIdx.x & 31;
    const int w    = threadIdx.x >> 5;
    const int wq   = blockIdx.x * 8 + w;
    if (wq >= BB * NN) return;
    const int b = wq / NN, i = wq % NN;
    const int base = (w * 32 + lane) * KK;
    for (int t = 0; t < KK; ++t) { sd[base + t] = 3.4e38f; si[base + t] = i; }
    const float xi0 = x[(b * NN + i) * 3 + 0];
    const float xi1 = x[(b * NN + i) * 3 + 1];
    const float xi2 = x[(b * NN + i) * 3 + 2];
    for (int j = lane; j < NN; j += 32) {
        float d0 = x[(b * NN + j) * 3 + 0] - xi0;
        float d1 = x[(b * NN + j) * 3 + 1] - xi1;
        float d2 = x[(b * NN + j) * 3 + 2] - xi2;
        float d  = d0 * d0 + d1 * d1 + d2 * d2;
        if (d < sd[base + KK - 1]) {
            int p = KK - 1;
            while (p > 0 && sd[base + p - 1] > d) {
                sd[base + p] = sd[base + p - 1];
                si[base + p] = si[base + p - 1];
                --p;
            }
            sd[base + p] = d; si[base + p] = j;
        }
    }
    // k-way merge of 32 sorted lists
    int ptr = 0;
    float cand = sd[base]; int ci = si[base];
    for (int r = 0; r < KK; ++r) {
        float m = cand;
        for (int off = 16; off; off >>= 1) m = fminf(m, __shfl_xor(m, off));
        unsigned long long bal = __ballot(cand == m);
        int wl = (int)__ffsll(bal) - 1;
        int widx = __shfl(ci, wl);
        if (lane == 0) idxOut[(b * NN + i) * KK + r] = widx;
        if (lane == wl) {
            ++ptr;
            cand = (ptr < KK) ? sd[base + ptr] : 3.4e38f;
            ci   = (ptr < KK) ? si[base + ptr] : i;
        }
    }
}

// ============================ 2. block1 (C_in = 3) ==========================
// Edge features: f = [nbr-ctr, ctr, cross(nbr,ctr)], p_c = W1[c,:]·f.
// Stats pass: per-channel sum / sumsq of ||p|| over (B,N,K).
// Block = 64 channels x 4 k-slots, 8 points of coords staged in LDS.
__global__ void vn1_stats_kernel(const float* __restrict__ x, const int* __restrict__ idx,
                                 const float* __restrict__ W1,
                                 float* __restrict__ ssum, float* __restrict__ ssq) {
    __shared__ float pc[8][21][3];
    __shared__ float rs[64], rq[64];
    const int b  = blockIdx.x / (NN / 8);
    const int n0 = (blockIdx.x % (NN / 8)) * 8;
    const int t  = threadIdx.x;
    for (int u = t; u < 8 * 21; u += 256) {
        int nn = u / 21, p = u % 21;
        int n  = n0 + nn;
        int j  = (p == 0) ? n : idx[(b * NN + n) * KK + (p - 1)];
        pc[nn][p][0] = x[(b * NN + j) * 3 + 0];
        pc[nn][p][1] = x[(b * NN + j) * 3 + 1];
        pc[nn][p][2] = x[(b * NN + j) * 3 + 2];
    }
    if (t < 64) { rs[t] = 0.f; rq[t] = 0.f; }
    __syncthreads();
    const int c = t & 63, slot = t >> 6;
    const float w0 = W1[c * 3 + 0], w1 = W1[c * 3 + 1], w2 = W1[c * 3 + 2];
    float ls = 0.f, lq = 0.f;
    for (int nn = 0; nn < 8; ++nn) {
        float c0 = pc[nn][0][0], c1 = pc[nn][0][1], c2 = pc[nn][0][2];
        for (int k = slot; k < KK; k += 4) {
            float a0 = pc[nn][1 + k][0], a1 = pc[nn][1 + k][1], a2 = pc[nn][1 + k][2];
            float e0 = a0 - c0, e1 = a1 - c1, e2 = a2 - c2;
            float x0 = a1 * c2 - a2 * c1, x1 = a2 * c0 - a0 * c2, x2 = a0 * c1 - a1 * c0;
            float p0 = w0 * e0 + w1 * c0 + w2 * x0;
            float p1 = w0 * e1 + w1 * c1 + w2 * x1;
            float p2 = w0 * e2 + w1 * c2 + w2 * x2;
            float nrm = sqrtf(p0 * p0 + p1 * p1 + p2 * p2) + EPSV;
            ls += nrm; lq += nrm * nrm;
        }
    }
    atomicAdd(&rs[c], ls); atomicAdd(&rq[c], lq);
    __syncthreads();
    if (t < 64) { atomicAdd(&ssum[t], rs[t]); atomicAdd(&ssq[t], rq[t]); }
}

__global__ void bn_finalize_kernel(const float* __restrict__ ssum, const float* __restrict__ ssq,
                                   float* __restrict__ mean, float* __restrict__ inv,
                                   int C, float cnt) {
    int c = blockIdx.x * 256 + threadIdx.x;
    if (c < C) {
        float m = ssum[c] / cnt;
        float v = ssq[c] / cnt - m * m;
        mean[c] = m;
        inv[c]  = rsqrtf(v + BNEPS);
    }
}

// Apply BN + VN-leaky (slope 0) and pool (mean over K). Output X[c][(b*3+v)*NN+n].
__global__ void vn1_apply_pool_kernel(const float* __restrict__ x, const int* __restrict__ idx,
                                      const float* __restrict__ W1, const float* __restrict__ D1,
                                      const float* __restrict__ g, const float* __restrict__ be,
                                      const float* __restrict__ mean, const float* __restrict__ inv,
                                      float* __restrict__ X) {
    __shared__ float pc[21][3];
    const int bn = blockIdx.x;
    const int b = bn / NN, n = bn % NN;
    const int c = threadIdx.x;
    for (int u = c; u < 21; u += 64) {
        int j = (u == 0) ? n : idx[bn * KK + (u - 1)];
        pc[u][0] = x[(b * NN + j) * 3 + 0];
        pc[u][1] = x[(b * NN + j) * 3 + 1];
        pc[u][2] = x[(b * NN + j) * 3 + 2];
    }
    __syncthreads();
    const float w0 = W1[c * 3], w1 = W1[c * 3 + 1], w2 = W1[c * 3 + 2];
    const float d0 = D1[c * 3], d1 = D1[c * 3 + 1], d2 = D1[c * 3 + 2];
    const float mn = mean[c], iv = inv[c], gg = g[c], bb = be[c];
    float a0 = 0.f, a1 = 0.f, a2 = 0.f;
    const float c0 = pc[0][0], c1 = pc[0][1], c2 = pc[0][2];
    for (int k = 0; k < KK; ++k) {
        float u0 = pc[1 + k][0], u1 = pc[1 + k][1], u2 = pc[1 + k][2];
        float e0 = u0 - c0, e1 = u1 - c1, e2 = u2 - c2;
        float x0 = u1 * c2 - u2 * c1, x1 = u2 * c0 - u0 * c2, x2 = u0 * c1 - u1 * c0;
        float p0 = w0 * e0 + w1 * c0 + w2 * x0;
        float p1 = w0 * e1 + w1 * c1 + w2 * x1;
        float p2 = w0 * e2 + w1 * c2 + w2 * x2;
        float q0 = d0 * e0 + d1 * c0 + d2 * x0;
        float q1 = d0 * e1 + d1 * c1 + d2 * x1;
        float q2 = d0 * e2 + d1 * c2 + d2 * x2;
        float nrm = sqrtf(p0 * p0 + p1 * p1 + p2 * p2) + EPSV;
        float s = (gg * (nrm - mn) * iv + bb) / nrm;
        p0 *= s; p1 *= s; p2 *= s;
        float dot = p0 * q0 + p1 * q1 + p2 * q2;
        if (dot < 0.f) {
            float f = dot / (q0 * q0 + q1 * q1 + q2 * q2 + EPSV);
            p0 -= f * q0; p1 -= f * q1; p2 -= f * q2;
        }
        a0 += p0; a1 += p1; a2 += p2;
    }
    const float s = 1.f / KK;
    X[(size_t)c * MTOT + (b * 3 + 0) * NN + n] = a0 * s;
    X[(size_t)c * MTOT + (b * 3 + 1) * NN + n] = a1 * s;
    X[(size_t)c * MTOT + (b * 3 + 2) * NN + n] = a2 * s;
}

// ============================ 3. WMMA dual GEMM =============================
// P = W·X, Q = D·X with X:[Cin][MTOT]. One wave per 16(co) x 16(m) tile,
// K-loop in steps of 4 using V_WMMA_F32_16X16X4_F32; B operand shared by the
// two accumulators. A layout: lane=(M=lane%16, K=(lane/16)*2+r).
__global__ void gemm_dual_wmma_kernel(const float* __restrict__ W, const float* __restrict__ D,
                                      const float* __restrict__ X,
                                      float* __restrict__ P, float* __restrict__ Q,
                                      int Cin, int Cout) {
    const int lane = threadIdx.x & 31;
    const int wv   = blockIdx.x * 8 + (threadIdx.x >> 5);
    const int ntc  = Cout >> 4;
    const int ntm  = MTOT >> 4;
    if (wv >= ntc * ntm) return;
    const int co0 = (wv % ntc) << 4;
    const int m0  = (wv / ntc) << 4;
    const int mr  = lane & 15, half = lane >> 4;
    v8f accP = {}; v8f accQ = {};
    for (int k0 = 0; k0 < Cin; k0 += 4) {
        const int kb = k0 + half * 2;
        v2f a, d, bb;
        a.x  = W[(co0 + mr) * Cin + kb];
        a.y  = W[(co0 + mr) * Cin + kb + 1];
        d.x  = D[(co0 + mr) * Cin + kb];
        d.y  = D[(co0 + mr) * Cin + kb + 1];
        bb.x = X[(size_t)(kb + 0) * MTOT + m0 + mr];
        bb.y = X[(size_t)(kb + 1) * MTOT + m0 + mr];
        accP = __builtin_amdgcn_wmma_f32_16x16x4_f32(false, a, false, bb, (short)0, accP, false, false);
        accQ = __builtin_amdgcn_wmma_f32_16x16x4_f32(false, d, false, bb, (short)0, accQ, false, false);
    }
    for (int r = 0; r < 8; ++r) {
        P[(size_t)(co0 + r + half * 8) * MTOT + m0 + mr] = accP[r];
        Q[(size_t)(co0 + r + half * 8) * MTOT + m0 + mr] = accQ[r];
    }
}

// Per-channel norm statistics over (B,N) of P[C][(b*3+v)*NN+n].
__global__ void stats_kernel(const float* __restrict__ P,
                             float* __restrict__ ssum, float* __restrict__ ssq) {
    const int c  = blockIdx.y;
    const int t  = threadIdx.x;
    const int bn = blockIdx.x * 256 + t;
    const int b = bn / NN, n = bn % NN;
    const float* base = P + (size_t)c * MTOT + (size_t)b * 3 * NN + n;
    float p0 = base[0], p1 = base[NN], p2 = base[2 * NN];
    float nrm = sqrtf(p0 * p0 + p1 * p1 + p2 * p2) + EPSV;
    float ls = nrm, lq = nrm * nrm;
    for (int off = 16; off; off >>= 1) { ls += __shfl_xor(ls, off); lq += __shfl_xor(lq, off); }
    __shared__ float rs[8], rq[8];
    if ((t & 31) == 0) { rs[t >> 5] = ls; rq[t >> 5] = lq; }
    __syncthreads();
    if (t == 0) {
        float S = 0.f, Sq = 0.f;
        for (int i = 0; i < 8; ++i) { S += rs[i]; Sq += rq[i]; }
        atomicAdd(&ssum[c], S); atomicAdd(&ssq[c], Sq);
    }
}

// BN scale + VN leaky, in place into P (becomes next layer's X).
__global__ void apply_leaky_kernel(float* __restrict__ P, const float* __restrict__ Q,
                                   const float* __restrict__ g, const float* __restrict__ be,
                                   const float* __restrict__ mean, const float* __restrict__ inv,
                                   int C) {
    int id = blockIdx.x * 256 + threadIdx.x;
    if (id >= C * BB * NN) return;
    int c  = id / (BB * NN);
    int bn = id % (BB * NN);
    int b = bn / NN, n = bn % NN;
    size_t o = (size_t)c * MTOT + (size_t)b * 3 * NN + n;
    float p0 = P[o], p1 = P[o + NN], p2 = P[o + 2 * NN];
    float q0 = Q[o], q1 = Q[o + NN], q2 = Q[o + 2 * NN];
    float nrm = sqrtf(p0 * p0 + p1 * p1 + p2 * p2) + EPSV;
    float s = (g[c] * (nrm - mean[c]) * inv[c] + be[c]) / nrm;
    p0 *= s; p1 *= s; p2 *= s;
    float dot = p0 * q0 + p1 * q1 + p2 * q2;
    if (dot < 0.f) {
        float f = dot / (q0 * q0 + q1 * q1 + q2 * q2 + EPSV);
        p0 -= f * q0; p1 -= f * q1; p2 -= f * q2;
    }
    P[o] = p0; P[o + NN] = p1; P[o + 2 * NN] = p2;
}

// ====================== 4. conv5 + BN5 + mean over N ========================
// Two WMMA passes, never materializing the 201MB [1024,3,B,N] tensor.
// Each wave: 16(co) x 16(n) tile for one b, 3 accumulators (v=0,1,2) sharing
// the A (weight) operand, so norms are elementwise across accumulators.
__global__ void conv5_stats_wmma_kernel(const float* __restrict__ W5, const float* __restrict__ X,
                                        float* __restrict__ ssum, float* __restrict__ ssq) {
    const int lane = threadIdx.x & 31;
    const int wv   = blockIdx.x * 8 + (threadIdx.x >> 5);
    const int ntc  = 1024 / 16, ntn = NN / 16;
    if (wv >= ntc * BB * ntn) return;
    const int co0 = (wv % ntc) << 4;
    const int rem = wv / ntc;
    const int b   = rem / ntn;
    const int n0  = (rem % ntn) << 4;
    const int mr  = lane & 15, half = lane >> 4;
    v8f a0 = {}, a1 = {}, a2 = {};
    for (int k0 = 0; k0 < 128; k0 += 4) {
        const int kb = k0 + half * 2;
        v2f a; a.x = W5[(co0 + mr) * 128 + kb]; a.y = W5[(co0 + mr) * 128 + kb + 1];
        v2f b0, b1, b2;
        b0.x = X[(size_t)(kb + 0) * MTOT + (b * 3 + 0) * NN + n0 + mr];
        b0.y = X[(size_t)(kb + 1) * MTOT + (b * 3 + 0) * NN + n0 + mr];
        b1.x = X[(size_t)(kb + 0) * MTOT + (b * 3 + 1) * NN + n0 + mr];
        b1.y = X[(size_t)(kb + 1) * MTOT + (b * 3 + 1) * NN + n0 + mr];
        b2.x = X[(size_t)(kb + 0) * MTOT + (b * 3 + 2) * NN + n0 + mr];
        b2.y = X[(size_t)(kb + 1) * MTOT + (b * 3 + 2) * NN + n0 + mr];
        a0 = __builtin_amdgcn_wmma_f32_16x16x4_f32(false, a, false, b0, (short)0, a0, false, false);
        a1 = __builtin_amdgcn_wmma_f32_16x16x4_f32(false, a, false, b1, (short)0, a1, false, false);
        a2 = __builtin_amdgcn_wmma_f32_16x16x4_f32(false, a, false, b2, (short)0, a2, false, false);
    }
    for (int r = 0; r < 8; ++r) {
        float nrm = sqrtf(a0[r] * a0[r] + a1[r] * a1[r] + a2[r] * a2[r]) + EPSV;
        float ls = nrm, lq = nrm * nrm;
        for (int off = 1; off <= 8; off <<= 1) { ls += __shfl_xor(ls, off); lq += __shfl_xor(lq, off); }
        if (mr == 0) {
            int c = co0 + r + half * 8;
            atomicAdd(&ssum[c], ls); atomicAdd(&ssq[c], lq);
        }
    }
}

__global__ void conv5_apply_wmma_kernel(const float* __restrict__ W5, const float* __restrict__ X,
                                        const float* __restrict__ g, const float* __restrict__ be,
                                        const float* __restrict__ mean, const float* __restrict__ inv,
                                        float* __restrict__ out) {
    const int lane = threadIdx.x & 31;
    const int wv   = blockIdx.x * 8 + (threadIdx.x >> 5);
    const int ntc  = 1024 / 16, ntn = NN / 16;
    if (wv >= ntc * BB * ntn) return;
    const int co0 = (wv % ntc) << 4;
    const int rem = wv / ntc;
    const int b   = rem / ntn;
    const int n0  = (rem % ntn) << 4;
    const int mr  = lane & 15, half = lane >> 4;
    v8f a0 = {}, a1 = {}, a2 = {};
    for (int k0 = 0; k0 < 128; k0 += 4) {
        const int kb = k0 + half * 2;
        v2f a; a.x = W5[(co0 + mr) * 128 + kb]; a.y = W5[(co0 + mr) * 128 + kb + 1];
        v2f b0, b1, b2;
        b0.x = X[(size_t)(kb + 0) * MTOT + (b * 3 + 0) * NN + n0 + mr];
        b0.y = X[(size_t)(kb + 1) * MTOT + (b * 3 + 0) * NN + n0 + mr];
        b1.x = X[(size_t)(kb + 0) * MTOT + (b * 3 + 1) * NN + n0 + mr];
        b1.y = X[(size_t)(kb + 1) * MTOT + (b * 3 + 1) * NN + n0 + mr];
        b2.x = X[(size_t)(kb + 0) * MTOT + (b * 3 + 2) * NN + n0 + mr];
        b2.y = X[(size_t)(kb + 1) * MTOT + (b * 3 + 2) * NN + n0 + mr];
        a0 = __builtin_amdgcn_wmma_f32_16x16x4_f32(false, a, false, b0, (short)0, a0, false, false);
        a1 = __builtin_amdgcn_wmma_f32_16x16x4_f32(false, a, false, b1, (short)0, a1, false, false);
        a2 = __builtin_amdgcn_wmma_f32_16x16x4_f32(false, a, false, b2, (short)0, a2, false, false);
    }
    for (int r = 0; r < 8; ++r) {
        const int c = co0 + r + half * 8;
        float nrm = sqrtf(a0[r] * a0[r] + a1[r] * a1[r] + a2[r] * a2[r]) + EPSV;
        float s = (g[c] * (nrm - mean[c]) * inv[c] + be[c]) / nrm;
        float v0 = a0[r] * s, v1 = a1[r] * s, v2 = a2[r] * s;
        for (int off = 1; off <= 8; off <<= 1) {
            v0 += __shfl_xor(v0, off); v1 += __shfl_xor(v1, off); v2 += __shfl_xor(v2, off);
        }
        if (mr == 0) {
            const float sc = 1.f / NN;
            atomicAdd(&out[((size_t)b * 1024 + c) * 3 + 0], v0 * sc);
            atomicAdd(&out[((size_t)b * 1024 + c) * 3 + 1], v1 * sc);
            atomicAdd(&out[((size_t)b * 1024 + c) * 3 + 2], v2 * sc);
        }
    }
}

// ============================ host orchestration ============================
extern "C" void kernel_launch(void* const* d_in, const int* in_sizes, int n_in,
                              void* d_out, int out_size, void* d_ws, size_t ws_size,
                              hipStream_t stream) {
    (void)in_sizes; (void)n_in; (void)out_size; (void)ws_size;
    const float* x  = (const float*)d_in[0];
    const float* W1 = (const float*)d_in[1];
    const float* D1 = (const float*)d_in[2];
    const float* g1 = (const float*)d_in[3];
    const float* b1 = (const float*)d_in[4];
    const float* Ws[3] = {(const float*)d_in[5], (const float*)d_in[9],  (const float*)d_in[13]};
    const float* Ds[3] = {(const float*)d_in[6], (const float*)d_in[10], (const float*)d_in[14]};
    const float* gs[3] = {(const float*)d_in[7], (const float*)d_in[11], (const float*)d_in[15]};
    const float* bs[3] = {(const float*)d_in[8], (const float*)d_in[12], (const float*)d_in[16]};
    const float* W5 = (const float*)d_in[17];
    const float* g5 = (const float*)d_in[18];
    const float* b5 = (const float*)d_in[19];
    float* out = (float*)d_out;

    // workspace layout (~77 MB)
    char* ws = (char*)d_ws;
    int*  idx = (int*)ws;
    size_t o = ((size_t)BB * NN * KK * sizeof(int) + 255) & ~(size_t)255;
    float* bufX = (float*)(ws + o); o += (size_t)128 * MTOT * sizeof(float);
    float* bufP = (float*)(ws + o); o += (size_t)128 * MTOT * sizeof(float);
    float* bufQ = (float*)(ws + o); o += (size_t)128 * MTOT * sizeof(float);
    float* ssum = (float*)(ws + o); o += 1024 * sizeof(float);
    float* ssq  = (float*)(ws + o); o += 1024 * sizeof(float);
    float* smn  = (float*)(ws + o); o += 1024 * sizeof(float);
    float* sinv = (float*)(ws + o); o += 1024 * sizeof(float);

    // 1) kNN graph
    knn_kernel<<<(BB * NN) / 8, 256, 0, stream>>>(x, idx);

    // 2) block1: stats -> finalize -> apply + pool over K  (writes bufX, 64ch)
    zero_f32<<<8, 256, 0, stream>>>(ssum, 2048);   // ssum+ssq contiguous
    vn1_stats_kernel<<<BB * (NN / 8), 256, 0, stream>>>(x, idx, W1, ssum, ssq);
    bn_finalize_kernel<<<1, 256, 0, stream>>>(ssum, ssq, smn, sinv, 64, (float)(BB * NN * KK));
    vn1_apply_pool_kernel<<<BB * NN, 64, 0, stream>>>(x, idx, W1, D1, g1, b1, smn, sinv, bufX);

    // 3) blocks 2..4: dual WMMA GEMM -> stats -> finalize -> BN+leaky in place
    const int cins[3]  = {64, 64, 64};
    const int couts[3] = {64, 64, 128};
    float* Xc = bufX;
    float* Pc = bufP;
    for (int L = 0; L < 3; ++L) {
        const int Cin = cins[L], Cout = couts[L];
        const int tiles = (Cout / 16) * (MTOT / 16);
        gemm_dual_wmma_kernel<<<(tiles + 7) / 8, 256, 0, stream>>>(Ws[L], Ds[L], Xc, Pc, bufQ, Cin, Cout);
        zero_f32<<<8, 256, 0, stream>>>(ssum, 2048);
        dim3 sg(BB * NN / 256, Cout);
        stats_kernel<<<sg, 256, 0, stream>>>(Pc, ssum, ssq);
        bn_finalize_kernel<<<1, 256, 0, stream>>>(ssum, ssq, smn, sinv, Cout, (float)(BB * NN));
        const int tot = Cout * BB * NN;
        apply_leaky_kernel<<<(tot + 255) / 256, 256, 0, stream>>>(Pc, bufQ, gs[L], bs[L], smn, sinv, Cout);
        float* t = Xc; Xc = Pc; Pc = t;   // Pc (post-activation) becomes next input
    }

    // 4) conv5 + bn5 + mean over N: two WMMA passes, no 1024-ch tensor stored
    zero_f32<<<8, 256, 0, stream>>>(ssum, 2048);
    const int waves5 = (1024 / 16) * BB * (NN / 16);
    conv5_stats_wmma_kernel<<<(waves5 + 7) / 8, 256, 0, stream>>>(W5, Xc, ssum, ssq);
    bn_finalize_kernel<<<4, 256, 0, stream>>>(ssum, ssq, smn, sinv, 1024, (float)(BB * NN));
    zero_f32<<<(BB * 1024 * 3 + 255) / 256, 256, 0, stream>>>(out, BB * 1024 * 3);
    conv5_apply_wmma_kernel<<<(waves5 + 7) / 8, 256, 0, stream>>>(W5, Xc, g5, b5, smn, sinv, out);
}